// StegoEmbeddingLayer_12137577578660
// MI455X (gfx1250) — compile-verified
//
#include <hip/hip_runtime.h>

// out = floor(image * 0.5), elementwise over 25,165,824 float32 values.
// Pure HBM-streaming problem: 201.3 MB of traffic, ~8.6 us floor at 23.3 TB/s.
// Strategy: b128 non-temporal loads/stores (gfx1250 TH=NT cache policy),
// 4 x float4 per thread for deep VMEM clauses, wave32-friendly 256-thread blocks.

typedef __attribute__((ext_vector_type(4))) float float4v;

static constexpr int TPB = 256;  // 8 wave32 waves per block
static constexpr int VPT = 4;    // float4 vectors per thread (64 B per thread)

__global__ __launch_bounds__(TPB)
void stego_halve_vec_kernel(const float4v* __restrict__ in,
                            float4v* __restrict__ out,
                            int n4) {
    const int base = blockIdx.x * (TPB * VPT) + threadIdx.x;

    if (base + (VPT - 1) * TPB < n4) {
        // Fast path: all 4 vectors in range. Issue all loads first so the
        // compiler can clause 4 independent global_load_b128 th:NT ops.
        float4v v[VPT];
#pragma unroll
        for (int k = 0; k < VPT; ++k)
            v[k] = __builtin_nontemporal_load(&in[base + k * TPB]);
#pragma unroll
        for (int k = 0; k < VPT; ++k) {
            float4v r = __builtin_elementwise_floor(v[k] * 0.5f);
            __builtin_nontemporal_store(r, &out[base + k * TPB]);
        }
    } else {
        // Boundary block: per-vector guard.
#pragma unroll
        for (int k = 0; k < VPT; ++k) {
            const int i = base + k * TPB;
            if (i < n4) {
                float4v v = __builtin_nontemporal_load(&in[i]);
                float4v r = __builtin_elementwise_floor(v * 0.5f);
                __builtin_nontemporal_store(r, &out[i]);
            }
        }
    }
}

// Scalar tail for n % 4 != 0 (not hit for this problem size, kept for safety).
__global__ void stego_halve_tail_kernel(const float* __restrict__ in,
                                        float* __restrict__ out,
                                        int start, int n) {
    const int i = start + threadIdx.x;
    if (i < n) {
        out[i] = __builtin_floorf(in[i] * 0.5f);
    }
}

extern "C" void kernel_launch(void* const* d_in, const int* in_sizes, int n_in,
                              void* d_out, int out_size, void* d_ws, size_t ws_size,
                              hipStream_t stream) {
    (void)n_in; (void)d_ws; (void)ws_size;

    const float* image = (const float*)d_in[0];   // (8,3,1024,1024) float32
    // d_in[1] (text_bits) is mathematically dead in the reference.
    float* out = (float*)d_out;

    const int n  = in_sizes[0];      // 25,165,824
    const int n4 = n >> 2;           // float4 count
    const int rem = n & 3;

    if (n4 > 0) {
        const int elems_per_block = TPB * VPT;
        const int blocks = (n4 + elems_per_block - 1) / elems_per_block;
        stego_halve_vec_kernel<<<blocks, TPB, 0, stream>>>(
            (const float4v*)image, (float4v*)out, n4);
    }
    if (rem > 0) {
        stego_halve_tail_kernel<<<1, 32, 0, stream>>>(image, out, n4 << 2, n);
    }
    (void)out_size;
}